// LennardJonesForce_7473243095377
// MI455X (gfx1250) — compile-verified
//
#include <hip/hip_runtime.h>

// Lennard-Jones edge energy + forces (scatter-add) for gfx1250 / MI455X.
//
// d_in[0] = pos     [N_ATOMS*3] f32
// d_in[1] = epsilon [N_EDGES]   f32
// d_in[2] = sigma   [N_EDGES]   f32
// d_in[3] = src     [N_EDGES]   i32
// d_in[4] = dst     [N_EDGES]   i32
// d_out   = [ energy (1 f32) | forces (N_ATOMS*3 f32) ]

#define BOXF     30.0f
#define INV_BOXF (1.0f / 30.0f)
#define RC2F     100.0f   // cutoff^2

typedef float f4 __attribute__((ext_vector_type(4)));
typedef int   i4 __attribute__((ext_vector_type(4)));

__global__ void lj_zero_kernel(float* __restrict__ out, int n) {
    int i = blockIdx.x * blockDim.x + threadIdx.x;
    if (i < n) out[i] = 0.0f;
}

// One edge: returns energy contribution, scatter-adds forces.
__device__ __forceinline__ float lj_edge(const float* __restrict__ pos,
                                         float ex, float sg, int s, int t,
                                         float* __restrict__ forces)
{
    // Gather endpoints (pos is L2-resident: 600 KB << 192 MB L2) -> b96 loads.
    float dx = pos[3 * s + 0] - pos[3 * t + 0];
    float dy = pos[3 * s + 1] - pos[3 * t + 1];
    float dz = pos[3 * s + 2] - pos[3 * t + 2];

    // Minimum image (cubic PBC). rintf == round-half-even == jnp.round.
    dx -= BOXF * rintf(dx * INV_BOXF);
    dy -= BOXF * rintf(dy * INV_BOXF);
    dz -= BOXF * rintf(dz * INV_BOXF);

    const float r2 = dx * dx + dy * dy + dz * dz;
    if (r2 >= RC2F) return 0.0f;   // mask kills both energy and force grads

    const float m     = fmaxf(r2, 1e-12f);
    const float recip = 1.0f / m;          // single IEEE divide, reused twice
    const float inv   = (sg * sg) * recip; // (sigma/r)^2, guarded
    const float s6    = inv * inv * inv;   // (sigma/r)^6
    const float s12   = s6 * s6;           // (sigma/r)^12

    // f_src = -dE/dpos[src] = 24*eps*(2 s12 - s6)/m * d   (analytic grad)
    const float fc = 24.0f * ex * (2.0f * s12 - s6) * recip;
    const float fx = fc * dx;
    const float fy = fc * dy;
    const float fz = fc * dz;

    // Hardware FP32 global atomics (no-return -> STOREcnt path).
    unsafeAtomicAdd(&forces[3 * s + 0],  fx);
    unsafeAtomicAdd(&forces[3 * s + 1],  fy);
    unsafeAtomicAdd(&forces[3 * s + 2],  fz);
    unsafeAtomicAdd(&forces[3 * t + 0], -fx);
    unsafeAtomicAdd(&forces[3 * t + 1], -fy);
    unsafeAtomicAdd(&forces[3 * t + 2], -fz);

    return 4.0f * ex * (s12 - s6);
}

__global__ __launch_bounds__(256) void lj_force_kernel(
    const float* __restrict__ pos,
    const float* __restrict__ eps,
    const float* __restrict__ sig,
    const int*   __restrict__ src,
    const int*   __restrict__ dst,
    float*       __restrict__ energy,   // d_out + 0
    float*       __restrict__ forces,   // d_out + 1
    int nEdges)
{
    const int tid      = blockIdx.x * blockDim.x + threadIdx.x;
    const int nThreads = gridDim.x * blockDim.x;
    const int nTiles   = nEdges >> 2;          // 4 edges per tile

    const f4* __restrict__ eps4 = (const f4*)eps;
    const f4* __restrict__ sig4 = (const f4*)sig;
    const i4* __restrict__ src4 = (const i4*)src;
    const i4* __restrict__ dst4 = (const i4*)dst;

    float eAcc = 0.0f;

    // ---- vectorized main loop: 128-bit streaming loads, 8 gathers/iter ----
    // Grid is sized so this executes exactly once per thread for the target
    // problem size (perfect balance); the loop + prefetch remain for
    // generality on larger inputs.
    for (int t = tid; t < nTiles; t += nThreads) {
        if (t + nThreads < nTiles) {
            // global_prefetch_b8: pull next iteration's lines toward GL2.
            __builtin_prefetch(&eps4[t + nThreads], 0, 1);
            __builtin_prefetch(&sig4[t + nThreads], 0, 1);
            __builtin_prefetch(&src4[t + nThreads], 0, 1);
            __builtin_prefetch(&dst4[t + nThreads], 0, 1);
        }

        // Single-use streams: non-temporal b128 loads (TH=NT).
        const f4 ex = __builtin_nontemporal_load(&eps4[t]);
        const f4 sg = __builtin_nontemporal_load(&sig4[t]);
        const i4 sv = __builtin_nontemporal_load(&src4[t]);
        const i4 dv = __builtin_nontemporal_load(&dst4[t]);

        #pragma unroll
        for (int k = 0; k < 4; ++k)
            eAcc += lj_edge(pos, ex[k], sg[k], sv[k], dv[k], forces);
    }

    // ---- scalar tail (nEdges % 4) ----
    for (int e = (nEdges & ~3) + tid; e < nEdges; e += nThreads)
        eAcc += lj_edge(pos, eps[e], sig[e], src[e], dst[e], forces);

    // ---- energy: wave32 butterfly -> LDS -> one atomic per block ----
    #pragma unroll
    for (int m = 16; m >= 1; m >>= 1)
        eAcc += __shfl_xor(eAcc, m, 32);

    __shared__ float waveSum[8];   // 256 threads = 8 wave32
    const int lane = threadIdx.x & 31;
    const int wid  = threadIdx.x >> 5;
    if (lane == 0) waveSum[wid] = eAcc;
    __syncthreads();

    if (wid == 0) {
        float v = (lane < 8) ? waveSum[lane] : 0.0f;
        #pragma unroll
        for (int m = 4; m >= 1; m >>= 1)
            v += __shfl_xor(v, m, 32);
        if (lane == 0) unsafeAtomicAdd(energy, v);
    }
}

extern "C" void kernel_launch(void* const* d_in, const int* in_sizes, int n_in,
                              void* d_out, int out_size, void* d_ws, size_t ws_size,
                              hipStream_t stream) {
    const float* pos = (const float*)d_in[0];
    const float* eps = (const float*)d_in[1];
    const float* sig = (const float*)d_in[2];
    const int*   src = (const int*)d_in[3];
    const int*   dst = (const int*)d_in[4];

    const int nEdges = in_sizes[1];          // 3,200,000
    float* out = (float*)d_out;              // [0]=energy, [1..]=forces

    // Zero energy + forces (harness poisons d_out before timing).
    lj_zero_kernel<<<(out_size + 255) / 256, 256, 0, stream>>>(out, out_size);

    // Exactly one 4-edge tile per thread: 800K tiles -> 3125 blocks of 256.
    // 25K wave32s across the chip hide gather + atomic latency; perfectly
    // balanced (no second grid-stride iteration for the target size).
    const int threads = 256;
    const int nTiles  = nEdges >> 2;
    int blocks = (nTiles + threads - 1) / threads;
    if (blocks < 1)     blocks = 1;
    if (blocks > 16384) blocks = 16384;       // grid-stride kicks in beyond this
    lj_force_kernel<<<blocks, threads, 0, stream>>>(
        pos, eps, sig, src, dst, out, out + 1, nEdges);
}